// RaggedMoEScatterModule_62216896250152
// MI455X (gfx1250) — compile-verified
//
#include <hip/hip_runtime.h>
#include <hip/hip_bf16.h>

#define N_TOKENS  16384
#define TOP_K     2
#define N_EXPERTS 8
#define HIDDEN    2048
#define CAPACITY  4096
#define MOE_ROWS  (N_EXPERTS * CAPACITY)   // 32768 rows, == N_TOKENS*TOP_K (fully covered)

// The gfx1250 async global<->LDS builtins take pointers to 16-byte int vectors
// in AS(1) (global, printed as __device__) and AS(3) (LDS, printed as __shared__).
typedef int v4i __attribute__((vector_size(16)));
typedef __attribute__((address_space(1))) v4i* gv4i_p;
typedef __attribute__((address_space(3))) v4i* lv4i_p;

#define AS_G128(p) ((gv4i_p)(p))
#define AS_L128(p) ((lv4i_p)(p))

// One block per token: stage the 8KB activation row in LDS via the async DMA
// engine, then fan it out to both top-k expert slots. Every lane stores only
// the LDS bytes it loaded itself, so the only ordering needed is ASYNCcnt.
__global__ __launch_bounds__(256) void moe_scatter_async(
    const float* __restrict__ acts,
    const int*   __restrict__ mapped_slots,
    float*       __restrict__ out) {
  __shared__ float lds[HIDDEN];   // 8 KB stage (WGP has 320 KB LDS; trivial)

  const int t   = blockIdx.x;
  const int tid = threadIdx.x;    // 0..255

  const int s0 = mapped_slots[2 * t + 0];
  const int s1 = mapped_slots[2 * t + 1];

  const float* src  = acts + (size_t)t  * HIDDEN;
  float*       dst0 = out  + (size_t)s0 * HIDDEN;
  float*       dst1 = out  + (size_t)s1 * HIDDEN;

  // 256 lanes * 16B = 4KB per pass; two passes cover the 8KB row.
  const int o0 = tid * 4;          // float index, 16B-aligned
  const int o1 = 1024 + tid * 4;

  // global -> LDS (ASYNCcnt)
  __builtin_amdgcn_global_load_async_to_lds_b128(AS_G128(src + o0), AS_L128(lds + o0), 0, 0);
  __builtin_amdgcn_global_load_async_to_lds_b128(AS_G128(src + o1), AS_L128(lds + o1), 0, 0);

  asm volatile("s_wait_asynccnt 0x0" ::: "memory");

  // LDS -> global, two destinations (ASYNCcnt)
  __builtin_amdgcn_global_store_async_from_lds_b128(AS_G128(dst0 + o0), AS_L128(lds + o0), 0, 0);
  __builtin_amdgcn_global_store_async_from_lds_b128(AS_G128(dst0 + o1), AS_L128(lds + o1), 0, 0);
  __builtin_amdgcn_global_store_async_from_lds_b128(AS_G128(dst1 + o0), AS_L128(lds + o0), 0, 0);
  __builtin_amdgcn_global_store_async_from_lds_b128(AS_G128(dst1 + o1), AS_L128(lds + o1), 0, 0);

  asm volatile("s_wait_asynccnt 0x0" ::: "memory");
}

// Tuple outputs 2 and 3: expert_cumsum (8 i32) and mapped_slots (32768 i32),
// copied bit-exact into the tail of d_out after the moe_input block.
__global__ __launch_bounds__(256) void copy_tail(
    const int* __restrict__ expert_cumsum,
    const int* __restrict__ mapped_slots,
    int*       __restrict__ tail) {
  const int i = blockIdx.x * blockDim.x + threadIdx.x;
  if (i < N_EXPERTS) tail[i] = expert_cumsum[i];
  if (i < MOE_ROWS)  tail[N_EXPERTS + i] = mapped_slots[i];
}

extern "C" void kernel_launch(void* const* d_in, const int* in_sizes, int n_in,
                              void* d_out, int out_size, void* d_ws, size_t ws_size,
                              hipStream_t stream) {
  // setup_inputs() order: expert_cumsum, mapped_slots, activations,
  //                       expert_counts, assignments, offsets
  const int*   expert_cumsum = (const int*)d_in[0];
  const int*   mapped_slots  = (const int*)d_in[1];
  const float* activations   = (const float*)d_in[2];

  float* out = (float*)d_out;

  moe_scatter_async<<<N_TOKENS, 256, 0, stream>>>(activations, mapped_slots, out);

  const size_t moe_elems = (size_t)MOE_ROWS * HIDDEN;
  if ((size_t)out_size >= moe_elems + N_EXPERTS + MOE_ROWS) {
    int* tail = (int*)(out + moe_elems);
    copy_tail<<<(MOE_ROWS + 255) / 256, 256, 0, stream>>>(expert_cumsum, mapped_slots, tail);
  }
}